// VirtualNode_20048907337870
// MI455X (gfx1250) — compile-verified
//
#include <hip/hip_runtime.h>

#define N_NODES  262144
#define B_GRAPHS 1024
#define DIM      256

typedef float v2f __attribute__((ext_vector_type(2)));
typedef float v8f __attribute__((ext_vector_type(8)));

// ---------------------------------------------------------------------------
// Zero the workspace (sums + counts). Graph-capture safe, deterministic.
// ---------------------------------------------------------------------------
__global__ void vn_zero_ws(float* __restrict__ ws, int n) {
    int i = blockIdx.x * blockDim.x + threadIdx.x;
    if (i < n) ws[i] = 0.0f;
}

// ---------------------------------------------------------------------------
// Segment-sum over sorted batch ids.
// Block = 256 threads, owns 256 consecutive rows; thread t owns column t.
// Running register accumulator, flushed with a global atomic only when the
// graph id changes (batch is sorted -> ~#distinct-ids-per-chunk atomics).
// Thread 0 also accumulates per-graph counts the same way.
// ---------------------------------------------------------------------------
__global__ __launch_bounds__(256) void vn_segsum(const float* __restrict__ x,
                                                 const int*   __restrict__ batch,
                                                 float* __restrict__ sums,
                                                 float* __restrict__ counts) {
    __shared__ int sb[256];
    const int t    = threadIdx.x;
    const int row0 = blockIdx.x * 256;
    sb[t] = batch[row0 + t];
    __syncthreads();

    float acc = 0.0f;
    int   cur = sb[0];
    #pragma unroll 4
    for (int r = 0; r < 256; ++r) {
        const int g = sb[r];              // uniform across the wave
        if (g != cur) {
            atomicAdd(&sums[(size_t)cur * DIM + t], acc);
            acc = 0.0f;
            cur = g;
        }
        acc += x[(size_t)(row0 + r) * DIM + t];
    }
    atomicAdd(&sums[(size_t)cur * DIM + t], acc);

    if (t == 0) {
        int cg = sb[0];
        float c = 0.0f;
        for (int r = 0; r < 256; ++r) {
            const int g = sb[r];
            if (g != cg) { atomicAdd(&counts[cg], c); c = 0.0f; cg = g; }
            c += 1.0f;
        }
        atomicAdd(&counts[cg], c);
    }
}

// ---------------------------------------------------------------------------
// vn = (sums / max(counts,1)) @ W^T + bias, via V_WMMA_F32_16X16X4_F32.
// One wave computes one 16x16 output tile; K marched in steps of 4.
//   A (16x4 f32): lanes 0-15 -> M=lane,  v0=K0, v1=K1 ; lanes 16-31 -> K2,K3
//   B (4x16 f32): same K striping, N = lane & 15 ; B(k,n) = W[n*D + k]
//   C/D (16x16 f32): VGPR v -> M = v (+8 for upper lane half), N = lane & 15
// Grid: 1024 waves = 64 m-tiles x 16 n-tiles. EXEC is all-1s (no divergence).
// ---------------------------------------------------------------------------
__global__ __launch_bounds__(256) void vn_proj_wmma(const float* __restrict__ sums,
                                                    const float* __restrict__ counts,
                                                    const float* __restrict__ W,
                                                    const float* __restrict__ bias,
                                                    float* __restrict__ vn_out) {
    const int lane = threadIdx.x & 31;
    const int wave = blockIdx.x * (blockDim.x >> 5) + (threadIdx.x >> 5);
    const int tm   = (wave >> 4) << 4;     // 64 tiles along B_GRAPHS
    const int tn   = (wave & 15) << 4;     // 16 tiles along DIM
    const int l15  = lane & 15;
    const int kh   = (lane >> 4) << 1;     // 0 for lanes 0-15, 2 for 16-31

    const int arow = tm + l15;             // A row this lane supplies
    const int bcol = tn + l15;             // B column this lane supplies
    const float invc = 1.0f / fmaxf(counts[arow], 1.0f);

    v8f c;
    const float bn = bias[bcol];
    #pragma unroll
    for (int v = 0; v < 8; ++v) c[v] = bn;

    const float* __restrict__ ap = sums + (size_t)arow * DIM + kh;
    const float* __restrict__ bp = W    + (size_t)bcol * DIM + kh;

    #pragma unroll 4
    for (int k0 = 0; k0 < DIM; k0 += 4) {
        v2f a, b;
        a.x = ap[k0]     * invc;
        a.y = ap[k0 + 1] * invc;
        b.x = bp[k0];
        b.y = bp[k0 + 1];
        // 8 args: (neg_a, A, neg_b, B, c_mod, C, reuse_a, reuse_b)
        c = __builtin_amdgcn_wmma_f32_16x16x4_f32(false, a, false, b,
                                                  (short)0, c, false, false);
    }

    const int mbase = tm + ((lane >> 4) << 3);
    #pragma unroll
    for (int v = 0; v < 8; ++v) {
        vn_out[(size_t)(mbase + v) * DIM + bcol] = c[v];
    }
}

// ---------------------------------------------------------------------------
// x_out = x + vn[batch]   (float4 / 128-bit vector traffic)
// D/4 = 64 float4 per row; 64 consecutive threads share one batch[] load.
// vn (1 MB) is L2-resident, so gathers are cheap.
// ---------------------------------------------------------------------------
__global__ __launch_bounds__(256) void vn_residual(const float* __restrict__ x,
                                                   const int*   __restrict__ batch,
                                                   const float* __restrict__ vn,
                                                   float* __restrict__ x_out) {
    const size_t i   = (size_t)blockIdx.x * blockDim.x + threadIdx.x; // float4 idx
    const int    row = (int)(i >> 6);   // DIM/4 == 64
    const int    cv  = (int)(i & 63);
    const int    g   = batch[row];

    const float4 xv = ((const float4*)x)[i];
    const float4 vv = ((const float4*)vn)[(size_t)g * 64 + cv];
    float4 o;
    o.x = xv.x + vv.x;
    o.y = xv.y + vv.y;
    o.z = xv.z + vv.z;
    o.w = xv.w + vv.w;
    ((float4*)x_out)[i] = o;
}

// ---------------------------------------------------------------------------
extern "C" void kernel_launch(void* const* d_in, const int* in_sizes, int n_in,
                              void* d_out, int out_size, void* d_ws, size_t ws_size,
                              hipStream_t stream) {
    const float* x     = (const float*)d_in[0];   // [N, D] f32
    const int*   batch = (const int*)  d_in[1];   // [N]    i32 (sorted)
    const float* W     = (const float*)d_in[2];   // [D, D] f32
    const float* bias  = (const float*)d_in[3];   // [D]    f32
    // d_in[4] = layer_idx (unused)

    float* x_out  = (float*)d_out;                       // [N, D]
    float* vn_out = x_out + (size_t)N_NODES * DIM;       // [B, D]

    float* sums   = (float*)d_ws;                        // [B, D]
    float* counts = sums + (size_t)B_GRAPHS * DIM;       // [B]

    const int zn = B_GRAPHS * DIM + B_GRAPHS;
    vn_zero_ws<<<(zn + 255) / 256, 256, 0, stream>>>(sums, zn);

    vn_segsum<<<N_NODES / 256, 256, 0, stream>>>(x, batch, sums, counts);

    // 1024 waves = 128 blocks x 8 waves
    vn_proj_wmma<<<128, 256, 0, stream>>>(sums, counts, W, bias, vn_out);

    // N*D/4 float4 elements / 256 threads
    vn_residual<<<(N_NODES * (DIM / 4)) / 256, 256, 0, stream>>>(x, batch, vn_out, x_out);
}